// GAT_33285996544353
// MI455X (gfx1250) — compile-verified
//
#include <hip/hip_runtime.h>
#include <hip/hip_bf16.h>
#include <math.h>

typedef __attribute__((ext_vector_type(2))) float v2f;
typedef __attribute__((ext_vector_type(8))) float v8f;

#define NNODES 20000
#define NEDGES 320000
#define DIMC   768

// ---------------------------------------------------------------------------
// Fill kernel (init m=-inf, denom=0, acc=0)
// ---------------------------------------------------------------------------
__global__ void fill_kernel(float* __restrict__ p, float v, size_t n) {
    size_t i = (size_t)blockIdx.x * blockDim.x + threadIdx.x;
    size_t stride = (size_t)gridDim.x * blockDim.x;
    for (; i < n; i += stride) p[i] = v;
}

// ---------------------------------------------------------------------------
// GEMM  C[Nrows x 768] = X[Nrows x 768] @ W[768 x 768] + bias
// One wave computes a 16(M) x 64(N) strip of C via 4 f32 WMMA accumulators.
// Block = 256 threads = 8 waves = 128 rows x 64 cols.
// OOB rows are CLAMPED (not predicated) so the inner K-loop is branch-free;
// stores are guarded, so clamped rows never reach memory.
// ---------------------------------------------------------------------------
__global__ __launch_bounds__(256) void gemm_bias_wmma(
    const float* __restrict__ X, const float* __restrict__ W,
    const float* __restrict__ bias, float* __restrict__ C, int Nrows)
{
    const int lane = threadIdx.x & 31;
    const int wave = threadIdx.x >> 5;
    const int m0 = (blockIdx.x * 8 + wave) * 16;
    const int n0 = blockIdx.y * 64;

    // A fragment: lane l holds row M=l&15, K = klo,klo+1 (klo = l<16 ? 0 : 2)
    int mrow = m0 + (lane & 15);
    if (mrow >= Nrows) mrow = Nrows - 1;          // clamp: keeps loads unconditional
    const int klo  = (lane < 16) ? 0 : 2;
    const int ncol = n0 + (lane & 15);
    const float* xrow = X + (size_t)mrow * DIMC + klo;   // 8B aligned (klo even)
    const float* wcol = W + (size_t)klo * DIMC + ncol;

    v8f acc0 = {}, acc1 = {}, acc2 = {}, acc3 = {};

    for (int k = 0; k < DIMC; k += 4) {
        const v2f a = *(const v2f*)(xrow + k);           // global_load_b64
        const float* w0 = wcol + (size_t)k * DIMC;
        const float* w1 = w0 + DIMC;
        v2f b0 = { w0[0],  w1[0]  };
        v2f b1 = { w0[16], w1[16] };
        v2f b2 = { w0[32], w1[32] };
        v2f b3 = { w0[48], w1[48] };
        acc0 = __builtin_amdgcn_wmma_f32_16x16x4_f32(false, a, false, b0, (short)0, acc0, false, false);
        acc1 = __builtin_amdgcn_wmma_f32_16x16x4_f32(false, a, false, b1, (short)0, acc1, false, false);
        acc2 = __builtin_amdgcn_wmma_f32_16x16x4_f32(false, a, false, b2, (short)0, acc2, false, false);
        acc3 = __builtin_amdgcn_wmma_f32_16x16x4_f32(false, a, false, b3, (short)0, acc3, false, false);
    }

    // C/D layout: VGPR r -> row m0 + r + (lane<16 ? 0 : 8), col = group + (lane&15)
    const int mbase = m0 + ((lane < 16) ? 0 : 8);
    v8f accs[4] = {acc0, acc1, acc2, acc3};
#pragma unroll
    for (int j = 0; j < 4; ++j) {
        const int col = n0 + j * 16 + (lane & 15);
        const float bv = bias[col];
#pragma unroll
        for (int r = 0; r < 8; ++r) {
            const int row = mbase + r;
            if (row < Nrows) C[(size_t)row * DIMC + col] = accs[j][r] + bv;
        }
    }
}

// ---------------------------------------------------------------------------
// IEEE-ordered float atomic max via int/uint atomics
// ---------------------------------------------------------------------------
__device__ inline void atomicMaxFloat(float* addr, float val) {
    if (val >= 0.f) atomicMax((int*)addr, __float_as_int(val));
    else            atomicMin((unsigned int*)addr, __float_as_uint(val));
}

// ---------------------------------------------------------------------------
// Edge logits + segment max.  One wave per edge.
// logits[e,h] = sum_d leakyrelu(fs[src,h,d]+fd[dst,h,d], 0.2) * attn[h,d]
// ---------------------------------------------------------------------------
__global__ __launch_bounds__(256) void edge_logits_kernel(
    const float* __restrict__ fs, const float* __restrict__ fd,
    const float* __restrict__ attn, const int* __restrict__ src,
    const int* __restrict__ dst, float* __restrict__ logits,
    float* __restrict__ mbuf, int E, int H, int HD)
{
    const int wid  = threadIdx.x >> 5;
    const int lane = threadIdx.x & 31;
    const int e = blockIdx.x * (blockDim.x >> 5) + wid;
    if (e >= E) return;
    const int s = src[e], d = dst[e];
    const float* fsrow = fs + (size_t)s * DIMC;
    const float* fdrow = fd + (size_t)d * DIMC;
    for (int hh = 0; hh < H; ++hh) {
        const int base = hh * HD;
        float acc = 0.f;
        for (int c = lane; c < HD; c += 32) {
            float v = fsrow[base + c] + fdrow[base + c];
            v = (v > 0.f) ? v : 0.2f * v;
            acc += v * attn[base + c];
        }
#pragma unroll
        for (int off = 16; off > 0; off >>= 1) acc += __shfl_xor(acc, off, 32);
        if (lane == 0) {
            logits[(size_t)e * H + hh] = acc;
            atomicMaxFloat(&mbuf[(size_t)d * H + hh], acc);
        }
    }
}

// ---------------------------------------------------------------------------
// ex = exp(logit - m[dst]); denom[dst] += ex  (logits overwritten with ex)
// ---------------------------------------------------------------------------
__global__ void exp_denom_kernel(float* __restrict__ logits,
                                 const float* __restrict__ mbuf,
                                 float* __restrict__ denom,
                                 const int* __restrict__ dst, int E, int H)
{
    size_t idx = (size_t)blockIdx.x * blockDim.x + threadIdx.x;
    if (idx >= (size_t)E * H) return;
    const int e  = (int)(idx / H);
    const int hh = (int)(idx % H);
    const int d  = dst[e];
    const float ex = expf(logits[idx] - mbuf[(size_t)d * H + hh]);
    logits[idx] = ex;
    atomicAdd(&denom[(size_t)d * H + hh], ex);
}

// ---------------------------------------------------------------------------
// acc[dst,h,d] += (ex/denom[dst,h]) * fs[src,h,d].  One wave per edge.
// ---------------------------------------------------------------------------
__global__ __launch_bounds__(256) void scatter_kernel(
    const float* __restrict__ fs, const float* __restrict__ logits,
    const float* __restrict__ denom, const int* __restrict__ src,
    const int* __restrict__ dst, float* __restrict__ acc,
    int E, int H, int HD)
{
    const int wid  = threadIdx.x >> 5;
    const int lane = threadIdx.x & 31;
    const int e = blockIdx.x * (blockDim.x >> 5) + wid;
    if (e >= E) return;
    const int s = src[e], d = dst[e];
    const float* fsrow = fs + (size_t)s * DIMC;
    float* outrow = acc + (size_t)d * DIMC;
    for (int hh = 0; hh < H; ++hh) {
        const float alpha = logits[(size_t)e * H + hh] / denom[(size_t)d * H + hh];
        const int base = hh * HD;
        for (int c = lane; c < HD; c += 32)
            atomicAdd(&outrow[base + c], alpha * fsrow[base + c]);
    }
}

// ---------------------------------------------------------------------------
// out[row] = LayerNorm( elu(acc[row]) + (resid ? resid[row] : 0) ) * w + b
// One block (256 threads) per row; 768 = 3 elems/thread.
// ---------------------------------------------------------------------------
__global__ __launch_bounds__(256) void elu_residual_ln_kernel(
    const float* __restrict__ acc, const float* __restrict__ resid,
    const float* __restrict__ w, const float* __restrict__ b,
    float* __restrict__ out)
{
    __shared__ float ssum[256];
    __shared__ float ssq[256];
    const int row = blockIdx.x;
    const int tid = threadIdx.x;
    const float* arow = acc + (size_t)row * DIMC;

    float v[3];
    float s = 0.f, sq = 0.f;
#pragma unroll
    for (int i = 0; i < 3; ++i) {
        const int c = tid + i * 256;
        float a = arow[c];
        a = (a > 0.f) ? a : (expf(a) - 1.f);          // ELU(alpha=1)
        if (resid) a += resid[(size_t)row * DIMC + c];
        v[i] = a;
        s += a; sq += a * a;
    }
    ssum[tid] = s; ssq[tid] = sq;
    __syncthreads();
    for (int off = 128; off > 0; off >>= 1) {
        if (tid < off) { ssum[tid] += ssum[tid + off]; ssq[tid] += ssq[tid + off]; }
        __syncthreads();
    }
    const float mean = ssum[0] * (1.0f / DIMC);
    const float var  = ssq[0] * (1.0f / DIMC) - mean * mean;
    const float rstd = rsqrtf(var + 1e-5f);
#pragma unroll
    for (int i = 0; i < 3; ++i) {
        const int c = tid + i * 256;
        out[(size_t)row * DIMC + c] = (v[i] - mean) * rstd * w[c] + b[c];
    }
}

// ---------------------------------------------------------------------------
// Host launcher
// ---------------------------------------------------------------------------
extern "C" void kernel_launch(void* const* d_in, const int* in_sizes, int n_in,
                              void* d_out, int out_size, void* d_ws, size_t ws_size,
                              hipStream_t stream) {
    (void)in_sizes; (void)n_in; (void)out_size; (void)ws_size;

    const float* x     = (const float*)d_in[0];
    const int*   src   = (const int*)  d_in[1];
    const int*   dst   = (const int*)  d_in[2];
    const float* W1s   = (const float*)d_in[3];
    const float* b1s   = (const float*)d_in[4];
    const float* W1d   = (const float*)d_in[5];
    const float* b1d   = (const float*)d_in[6];
    const float* attn1 = (const float*)d_in[7];
    const float* ln1w  = (const float*)d_in[8];
    const float* ln1b  = (const float*)d_in[9];
    const float* Wos   = (const float*)d_in[10];
    const float* bos   = (const float*)d_in[11];
    const float* Wod   = (const float*)d_in[12];
    const float* bod   = (const float*)d_in[13];
    const float* attno = (const float*)d_in[14];
    const float* lnow  = (const float*)d_in[15];
    const float* lnob  = (const float*)d_in[16];
    float* out = (float*)d_out;

    const int N = NNODES, E = NEDGES, H1 = 8, HD1 = 96;
    const size_t ND = (size_t)N * DIMC;

    // Workspace layout (floats): fs | fd | acc | h | logits(E*8) | m(N*8) | denom(N*8)
    float* fs     = (float*)d_ws;
    float* fd     = fs + ND;
    float* accb   = fd + ND;
    float* h      = accb + ND;
    float* logits = h + ND;
    float* mbuf   = logits + (size_t)E * 8;
    float* denom  = mbuf + (size_t)N * 8;

    const dim3 gemmGrid((N + 127) / 128, DIMC / 64);
    const int edgeBlocks = (E + 7) / 8;   // 8 waves (edges) per 256-thread block

    // ---------------- Layer 1 (8 heads x 96) ----------------
    fill_kernel<<<1024, 256, 0, stream>>>(mbuf, -INFINITY, (size_t)N * H1);
    fill_kernel<<<1024, 256, 0, stream>>>(denom, 0.f, (size_t)N * H1);
    fill_kernel<<<2048, 256, 0, stream>>>(accb, 0.f, ND);

    gemm_bias_wmma<<<gemmGrid, 256, 0, stream>>>(x, W1s, b1s, fs, N);
    gemm_bias_wmma<<<gemmGrid, 256, 0, stream>>>(x, W1d, b1d, fd, N);

    edge_logits_kernel<<<edgeBlocks, 256, 0, stream>>>(fs, fd, attn1, src, dst,
                                                       logits, mbuf, E, H1, HD1);
    exp_denom_kernel<<<((size_t)E * H1 + 255) / 256, 256, 0, stream>>>(
        logits, mbuf, denom, dst, E, H1);
    scatter_kernel<<<edgeBlocks, 256, 0, stream>>>(fs, logits, denom, src, dst,
                                                   accb, E, H1, HD1);
    elu_residual_ln_kernel<<<N, 256, 0, stream>>>(accb, x, ln1w, ln1b, h);

    // ---------------- Layer 2 (1 head x 768) ----------------
    fill_kernel<<<256, 256, 0, stream>>>(mbuf, -INFINITY, (size_t)N);
    fill_kernel<<<256, 256, 0, stream>>>(denom, 0.f, (size_t)N);
    fill_kernel<<<2048, 256, 0, stream>>>(accb, 0.f, ND);

    gemm_bias_wmma<<<gemmGrid, 256, 0, stream>>>(h, Wos, bos, fs, N);
    gemm_bias_wmma<<<gemmGrid, 256, 0, stream>>>(h, Wod, bod, fd, N);

    edge_logits_kernel<<<edgeBlocks, 256, 0, stream>>>(fs, fd, attno, src, dst,
                                                       logits, mbuf, E, 1, DIMC);
    exp_denom_kernel<<<((size_t)E + 255) / 256, 256, 0, stream>>>(
        logits, mbuf, denom, dst, E, 1);
    scatter_kernel<<<edgeBlocks, 256, 0, stream>>>(fs, logits, denom, src, dst,
                                                   accb, E, 1, DIMC);
    elu_residual_ln_kernel<<<N, 256, 0, stream>>>(accb, nullptr, lnow, lnob, out);
}